// FluidVecSG_61718680043552
// MI455X (gfx1250) — compile-verified
//
#include <hip/hip_runtime.h>
#include <math.h>

#define DIM      300
#define BATCH    32
#define WCTX     8
#define NCH      4
#define NCO      8
#define NNEG     2392
#define FLAT_PER_B   (WCTX * NNEG)      // 19136
#define TILES_PER_B  (FLAT_PER_B / 16)  // 1196
#define NSPLIT   4
#define TILES_PER_BLK (TILES_PER_B / NSPLIT) // 299
#define NWAVES   4
#define TILE_FLOATS  (16 * DIM)         // 4800 floats = 19200 B per tile buffer
#define PAD_IDX  1
#define EPSV     1e-32f
#define NPART    (BATCH * NSPLIT + 1)

typedef __attribute__((ext_vector_type(2))) float        v2f;
typedef __attribute__((ext_vector_type(8))) float        v8f;
typedef __attribute__((ext_vector_type(4))) unsigned int u32x4;
typedef __attribute__((ext_vector_type(4))) int          i32x4;
typedef __attribute__((ext_vector_type(8))) int          i32x8;

// ---------------------------------------------------------------------------
// TDM gather: one descriptor pulls 16 indexed rows (300 f32 each) of word_emb
// into LDS at lds_off. 16-bit index gather mode (indices < 32000 fit).
// D# layout per CDNA5 ISA ch.8 (group0/1 bitfields, groups 2/3 = indices).
// clang-23 lane: 6-arg builtin, extra i32x8 (reserved/VADDR4) zeroed.
// ---------------------------------------------------------------------------
__device__ __forceinline__ void tdm_gather16(const float* __restrict__ word_emb,
                                             const int* __restrict__ idx_ptr,
                                             unsigned lds_off) {
  unsigned long long ga = (unsigned long long)(uintptr_t)word_emb;
  u32x4 g0 = {
      0x80000001u,                              // count=1 | gather_mode=1 | 16-bit idx
      lds_off,                                  // LDS byte offset of dest buffer
      (unsigned)(ga & 0xffffffffull),           // global_addr[31:0]
      (unsigned)((ga >> 32) & 0x01ffffffull) | 0x80000000u  // addr[56:32] | type=2
  };
  i32x8 g1 = {
      (int)0x00020000,      // workgroup_mask=0, data_size=2 (4 bytes)
      (int)(300u << 16),    // tensor_dim0[15:0]=300 in dw1[31:16]
      (int)(32000u << 16),  // tensor_dim0[31:16]=0 | tensor_dim1[15:0]=32000
      (int)(300u << 16),    // tensor_dim1[31:16]=0 | tile_dim0=300
      16,                   // tile_dim1 = #valid gather indices
      300,                  // tensor_dim0_stride[31:0] = 300 elements
      0, 0                  // stride hi / dim1_stride (ignored in gather mode)
  };
  int p[8];
#pragma unroll
  for (int j = 0; j < 8; j++) {
    unsigned lo = (unsigned)idx_ptr[2 * j]     & 0xffffu;
    unsigned hi = (unsigned)idx_ptr[2 * j + 1] & 0xffffu;
    p[j] = (int)(lo | (hi << 16));
  }
  i32x4 g2 = { p[0], p[1], p[2], p[3] };       // row_index_0..7
  i32x4 g3 = { p[4], p[5], p[6], p[7] };       // row_index_8..15
  i32x8 g4 = { 0, 0, 0, 0, 0, 0, 0, 0 };       // reserved / unused group
  __builtin_amdgcn_tensor_load_to_lds(g0, g1, g2, g3, g4, 0);
}

// ---------------------------------------------------------------------------
// Kernel 1: tgt[b,:] = masked char-bag + masked compo-bag            [32,300]
// ---------------------------------------------------------------------------
__global__ void __launch_bounds__(64) tgt_kernel(
    const float* __restrict__ char_emb, const float* __restrict__ compo_emb,
    const int* __restrict__ chars, const int* __restrict__ compos,
    float* __restrict__ tgt) {
  const int b = blockIdx.x;
  int ch[NCH], co[NCO];
#pragma unroll
  for (int i = 0; i < NCH; i++) ch[i] = chars[b * NCH + i];
#pragma unroll
  for (int i = 0; i < NCO; i++) co[i] = compos[b * NCO + i];
  for (int d = threadIdx.x; d < DIM; d += blockDim.x) {
    float s = 0.f;
#pragma unroll
    for (int i = 0; i < NCH; i++)
      if (ch[i] != PAD_IDX) s += char_emb[(size_t)ch[i] * DIM + d];
#pragma unroll
    for (int i = 0; i < NCO; i++)
      if (co[i] != PAD_IDX) s += compo_emb[(size_t)co[i] * DIM + d];
    tgt[b * DIM + d] = s;
  }
}

// ---------------------------------------------------------------------------
// Kernel 2: positive term  Σ_{b,w} log(sigmoid(tgt[b]·ctx[b,w]))
// ---------------------------------------------------------------------------
__global__ void __launch_bounds__(256) pos_kernel(
    const float* __restrict__ word_emb, const int* __restrict__ ctx_words,
    const float* __restrict__ tgt, float* __restrict__ partials) {
  __shared__ float red[256];
  const int t = threadIdx.x;
  const int b = t >> 3, w = t & 7;
  const int cw = ctx_words[b * WCTX + w];
  float s = 0.f;
  if (cw != PAD_IDX) {  // masked ctx word -> zero vector -> s = 0
    const float* cp = word_emb + (size_t)cw * DIM;
    const float* tp = tgt + b * DIM;
    for (int d = 0; d < DIM; d++) s += tp[d] * cp[d];
  }
  const float ls = (s >= 0.f) ? -log1pf(__expf(-s)) : (s - log1pf(__expf(s)));
  red[t] = ls;
  __syncthreads();
  if (t == 0) {
    float acc = 0.f;
    for (int i = 0; i < 256; i++) acc += red[i];
    partials[BATCH * NSPLIT] = acc;
  }
}

// ---------------------------------------------------------------------------
// Kernel 3: negative term. TDM gathers 16-row tiles into LDS (double-buffered
// per wave, TENSORcnt-synchronized); V_WMMA_F32_16X16X4_F32 marches K=300.
// grid = (32 batches, 4 tile-splits), 128 threads = 4 waves per block.
// ---------------------------------------------------------------------------
__global__ void __launch_bounds__(NWAVES * 32) neg_kernel(
    const float* __restrict__ word_emb, const int* __restrict__ noise,
    const float* __restrict__ tgt, float* __restrict__ partials) {
  extern __shared__ float dynls[];          // NWAVES * 2 * TILE_FLOATS
  __shared__ float tl[DIM];
  __shared__ float wsum[NWAVES];

  const int b   = blockIdx.x;
  const int q   = blockIdx.y;
  const int tid = threadIdx.x;

  for (int i = tid; i < DIM; i += NWAVES * 32) tl[i] = tgt[b * DIM + i];
  __syncthreads();

  const int wv   = __builtin_amdgcn_readfirstlane((int)(threadIdx.x >> 5));
  const int lane = tid & 31;
  const int m    = lane & 15;          // noise row within tile (A row)
  const int koff = (lane >> 4) << 1;   // 0 lanes 0-15, 2 lanes 16-31
  const int* nb  = noise + (size_t)b * FLAT_PER_B;
  const float* bptr = &tl[koff];

  float* buf0 = dynls + (size_t)wv * 2 * TILE_FLOATS;
  float* buf1 = buf0 + TILE_FLOATS;
  float* curbuf = buf0;  unsigned curoff = (unsigned)(uintptr_t)buf0;
  float* nxtbuf = buf1;  unsigned nxtoff = (unsigned)(uintptr_t)buf1;

  float acc = 0.f;
  int t = wv;
  if (t < TILES_PER_BLK)  // prime the pipeline
    tdm_gather16(word_emb, nb + (size_t)(q * TILES_PER_BLK + t) * 16, curoff);

  for (; t < TILES_PER_BLK; t += NWAVES) {   // uniform per wave
    const int tn = t + NWAVES;
    if (tn < TILES_PER_BLK) {
      // all ds reads of nxtbuf (previous round) must have landed before DMA
      asm volatile("s_wait_dscnt 0x0" ::: "memory");
      tdm_gather16(word_emb, nb + (size_t)(q * TILES_PER_BLK + tn) * 16, nxtoff);
      __builtin_amdgcn_s_wait_tensorcnt(1);  // oldest (curbuf) complete
    } else {
      __builtin_amdgcn_s_wait_tensorcnt(0);
    }
    asm volatile("" ::: "memory");           // no LDS reads above the wait

    const float* aptr = curbuf + m * DIM + koff;   // bank-conflict-free
    v8f d = {0.f, 0.f, 0.f, 0.f, 0.f, 0.f, 0.f, 0.f};
#pragma unroll 5
    for (int k0 = 0; k0 < DIM; k0 += 4) {
      v2f a  = *(const v2f*)(aptr + k0);     // ds_load_b64 (A tile)
      v2f bv = *(const v2f*)(bptr + k0);     // ds_load_b64 (tgt slice)
      d = __builtin_amdgcn_wmma_f32_16x16x4_f32(
          false, a, false, bv, (short)0, d, false, false);
    }
#pragma unroll
    for (int i = 0; i < 8; i++) {
      const float s = d[i];
      acc += __logf(__builtin_amdgcn_rcpf(1.0f + __expf(s)) + EPSV);
    }
    float* tb = curbuf; curbuf = nxtbuf; nxtbuf = tb;
    unsigned to = curoff; curoff = nxtoff; nxtoff = to;
  }

  for (int off = 16; off > 0; off >>= 1) acc += __shfl_xor(acc, off, 32);
  if (lane == 0) wsum[wv] = acc;
  __syncthreads();
  if (tid == 0) {
    float s = 0.f;
    for (int i = 0; i < NWAVES; i++) s += wsum[i];
    partials[b * NSPLIT + q] = s * (1.0f / 16.0f);  // undo 16-col replication
  }
}

// ---------------------------------------------------------------------------
// Kernel 4: deterministic finalize: out = -(pos + neg) / B
// ---------------------------------------------------------------------------
__global__ void fin_kernel(const float* __restrict__ partials,
                           float* __restrict__ out) {
  if (threadIdx.x == 0 && blockIdx.x == 0) {
    float s = 0.f;
    for (int i = 0; i < NPART; i++) s += partials[i];
    out[0] = -s / (float)BATCH;
  }
}

extern "C" void kernel_launch(void* const* d_in, const int* in_sizes, int n_in,
                              void* d_out, int out_size, void* d_ws, size_t ws_size,
                              hipStream_t stream) {
  const float* word_emb  = (const float*)d_in[0];
  const float* char_emb  = (const float*)d_in[1];
  const float* compo_emb = (const float*)d_in[2];
  const int*   chars     = (const int*)d_in[3];
  const int*   compos    = (const int*)d_in[4];
  const int*   ctx_words = (const int*)d_in[5];
  const int*   noise     = (const int*)d_in[6];

  float* ws       = (float*)d_ws;
  float* tgt      = ws;                 // 32*300 floats
  float* partials = ws + BATCH * DIM;   // NPART floats

  tgt_kernel<<<dim3(BATCH), dim3(64), 0, stream>>>(char_emb, compo_emb, chars,
                                                   compos, tgt);
  pos_kernel<<<dim3(1), dim3(256), 0, stream>>>(word_emb, ctx_words, tgt,
                                                partials);
  const size_t dyn_lds = (size_t)NWAVES * 2 * TILE_FLOATS * sizeof(float);
  neg_kernel<<<dim3(BATCH, NSPLIT), dim3(NWAVES * 32), dyn_lds, stream>>>(
      word_emb, noise, tgt, partials);
  fin_kernel<<<dim3(1), dim3(64), 0, stream>>>(partials, (float*)d_out);
}